// VolumeRenderer_74809740362339
// MI455X (gfx1250) — compile-verified
//
#include <hip/hip_runtime.h>

typedef __attribute__((ext_vector_type(2))) float v2f;
typedef __attribute__((ext_vector_type(8))) float v8f;

#define NS 128
#define MIN_ALPHA 1e-5f
#define MAX_ALPHA (1.0f - 1e-5f)
#define EPSV 1e-10f
#define INF_DIST 1e10f
#define WAVES 4          // waves per block
#define STRIDE 132       // 128 + 4 pad -> conflict-free LDS column reads

// One wave handles 16 rays. The exclusive cumprod (transmittance) is done as a
// log-space prefix sum Y = X * UpperTriOnes via chained V_WMMA_F32_16X16X4_F32,
// carry in the C operand, updated from D column 15 each tile.
__global__ __launch_bounds__(WAVES * 32) void volrender_wmma_scan(
    const float* __restrict__ sigma, const float* __restrict__ rgb,
    const float* __restrict__ z_vals, float* __restrict__ rgb_map,
    float* __restrict__ depth_map, float* __restrict__ acc_map,
    float* __restrict__ weights)
{
  __shared__ float Xs[WAVES][16 * STRIDE];

  const int lane = threadIdx.x & 31;
  const int wave = threadIdx.x >> 5;
  const int half = lane >> 4;      // 0: lanes 0-15, 1: lanes 16-31
  const int lrow = lane & 15;
  const int rayBase = (blockIdx.x * WAVES + wave) * 16;
  float* X = Xs[wave];

  // ---- Phase 1: coalesced streaming, X = log(1 - alpha + eps) into LDS ----
  for (int e = 0; e < 16; ++e) {
    const size_t rowOff = (size_t)(rayBase + e) * NS;
    const float4 s4 = ((const float4*)(sigma + rowOff))[lane];
    const float4 z4 = ((const float4*)(z_vals + rowOff))[lane];
    int nidx = lane * 4 + 4; nidx = (nidx > NS - 1) ? (NS - 1) : nidx;
    const float znext = z_vals[rowOff + nidx];

    float d[4], sg[4] = {s4.x, s4.y, s4.z, s4.w}, xs[4];
    d[0] = z4.y - z4.x;
    d[1] = z4.z - z4.y;
    d[2] = z4.w - z4.z;
    d[3] = (lane == 31) ? INF_DIST : (znext - z4.w);
#pragma unroll
    for (int k = 0; k < 4; ++k) {
      float a = 1.0f - __expf(-fmaxf(sg[k], 0.0f) * d[k]);
      a = fminf(fmaxf(a, MIN_ALPHA), MAX_ALPHA);
      xs[k] = __logf(1.0f - a + EPSV);
    }
    *(float4*)&X[e * STRIDE + lane * 4] = make_float4(xs[0], xs[1], xs[2], xs[3]);
  }
  __syncthreads();

  // ---- Phase 2/3: WMMA log-space scan + fused outputs ----
  float carry[8], accw[8], accd[8], accr[8], accg[8], accb[8];
#pragma unroll
  for (int i = 0; i < 8; ++i) {
    carry[i] = 0.f; accw[i] = 0.f; accd[i] = 0.f;
    accr[i] = 0.f;  accg[i] = 0.f; accb[i] = 0.f;
  }

  for (int t = 0; t < 8; ++t) {
    // C fragment: C[row = i + 8*half][col] = carry for that row (uniform per col)
    v8f acc;
#pragma unroll
    for (int i = 0; i < 8; ++i) acc[i] = carry[i];

    // X_t (16x16) * UT16 as 4 chained 16x16x4 f32 WMMAs.
    // A layout (f32 16x4): lanes 0-15 -> K=0,1 ; lanes 16-31 -> K=2,3 ; M = lane&15
    // B layout (f32 4x16): VGPR j, half h -> K = j + 2h ; N = lane&15
#pragma unroll
    for (int c = 0; c < 4; ++c) {
      const int k0 = c * 4 + half * 2;
      v2f a, b;
      a.x = X[lrow * STRIDE + t * 16 + k0];
      a.y = X[lrow * STRIDE + t * 16 + k0 + 1];
      b.x = (k0     <= lrow) ? 1.0f : 0.0f;   // UT16[k][n] = (k <= n)
      b.y = (k0 + 1 <= lrow) ? 1.0f : 0.0f;
      acc = __builtin_amdgcn_wmma_f32_16x16x4_f32(
          false, a, false, b, (short)0, acc, false, false);
    }

    // D layout: row = i + 8*half, col = lrow  -> sample s = 16t + lrow
    const int s = t * 16 + lrow;
#pragma unroll
    for (int i = 0; i < 8; ++i) {
      const int rloc = i + half * 8;
      const size_t roff = (size_t)(rayBase + rloc) * NS + s;
      const float x  = X[rloc * STRIDE + s];
      const float al = 1.0f + EPSV - __expf(x);     // recover clamped alpha
      const float T  = __expf(acc[i] - x);          // exclusive transmittance
      const float w  = al * T;
      weights[roff] = w;
      const float zv = z_vals[roff];
      const float* rp = rgb + roff * 3;
      accw[i] += w;
      accd[i] += w * zv;
      accr[i] += w * rp[0];
      accg[i] += w * rp[1];
      accb[i] += w * rp[2];
    }

    // carry = inclusive sum through this tile = D column 15 (lane 15 / 31)
#pragma unroll
    for (int i = 0; i < 8; ++i)
      carry[i] = __shfl(acc[i], half * 16 + 15, 32);
  }

  // ---- Per-row reductions across the 16 columns of each half-wave ----
#pragma unroll
  for (int i = 0; i < 8; ++i) {
    float w = accw[i], dd = accd[i], rr = accr[i], gg = accg[i], bb = accb[i];
#pragma unroll
    for (int m = 8; m >= 1; m >>= 1) {
      w  += __shfl_xor(w, m, 32);
      dd += __shfl_xor(dd, m, 32);
      rr += __shfl_xor(rr, m, 32);
      gg += __shfl_xor(gg, m, 32);
      bb += __shfl_xor(bb, m, 32);
    }
    if (lrow == 0) {
      const int r = rayBase + i + half * 8;
      const float bg = 1.0f - w;                    // white background
      rgb_map[r * 3 + 0] = rr + bg;
      rgb_map[r * 3 + 1] = gg + bg;
      rgb_map[r * 3 + 2] = bb + bg;
      depth_map[r] = dd;
      acc_map[r]   = w;
    }
  }
}

extern "C" void kernel_launch(void* const* d_in, const int* in_sizes, int n_in,
                              void* d_out, int out_size, void* d_ws, size_t ws_size,
                              hipStream_t stream) {
  const float* sigma = (const float*)d_in[0];   // [N, 128]
  const float* rgb   = (const float*)d_in[1];   // [N, 128, 3]
  const float* z     = (const float*)d_in[2];   // [N, 128]
  const int n_rays = in_sizes[0] / NS;

  // Output layout: rgb_map [N,3] | depth [N] | acc [N] | weights [N,128]
  float* out      = (float*)d_out;
  float* rgb_map  = out;
  float* depth    = out + (size_t)n_rays * 3;
  float* accm     = out + (size_t)n_rays * 4;
  float* weights  = out + (size_t)n_rays * 5;

  const int blocks = n_rays / (16 * WAVES);     // 1024 for N=65536
  volrender_wmma_scan<<<blocks, WAVES * 32, 0, stream>>>(
      sigma, rgb, z, rgb_map, depth, accm, weights);
}